// RMILoss_78554951844164
// MI455X (gfx1250) — compile-verified
//
#include <hip/hip_runtime.h>
#include <math.h>

// Problem constants (from reference)
#define NB     4
#define NC     21
#define H      512
#define W      512
#define PH     171          // pooled height: floor((512+2-3)/3)+1
#define PW     171
#define NH     169          // PH - (RADIUS-1)
#define NW     169
#define MTOT   (NH*NW)      // 28561 samples per (n,c)
#define NCHAN  (NB*NC)      // 84
#define POS_ALPHA 5e-4f
#define CLIP_MIN  1e-6f

// Gram kernel tiling: rows of the 169x169 crop grid per LDS tile
#define TROWS  32
#define LROWS  (TROWS + 2)          // staged pooled rows incl. 2-row halo
#define LDSN   (LROWS * PW)         // 34*171 = 5814 floats per plane

typedef __attribute__((ext_vector_type(2))) float v2f;
typedef __attribute__((ext_vector_type(8))) float v8f;

#if defined(__gfx1250__) && __has_builtin(__builtin_amdgcn_global_load_async_to_lds_b32)
#define USE_ASYNC_LDS 1
#else
#define USE_ASYNC_LDS 0
#endif

// Workspace layout (in floats):
//  [0] bce_sum  [1] valid_pixels  [2] rmi_acc  [3..15] pad
//  [WS_POOL_LA) pooled one-hot labels  : NCHAN*PH*PW
//  [WS_POOL_PR) pooled probs           : NCHAN*PH*PW
//  [WS_G)       raw Grams              : NCHAN*3*81  (ll, pp, lp)
//  [WS_S)       row sums               : NCHAN*2*9   (la, pr)
#define WS_POOL_LA 16
#define POOLSZ     (NCHAN*PH*PW)
#define WS_POOL_PR (WS_POOL_LA + POOLSZ)
#define GSZ        (NCHAN*3*81)
#define WS_G       (WS_POOL_PR + POOLSZ)
#define SSZ        (NCHAN*2*9)
#define WS_S       (WS_G + GSZ)

// ---------------------------------------------------------------- zero init
__global__ void rmi_zero_kernel(float* __restrict__ ws) {
    int i = blockIdx.x * blockDim.x + threadIdx.x;
    if (i < 16)                    ws[i] = 0.0f;
    else if (i < 16 + GSZ + SSZ)   ws[WS_G + (i - 16)] = 0.0f;
}

// ---------------- fused single-pass: BCE-with-logits sum + valid count +
//                  sigmoid/one-hot 3x3 stride-3 max-pool (pad 1).
// stride==kernel => every input pixel belongs to exactly one window, so this
// is the only pass over the 88MB logits tensor.
__global__ void __launch_bounds__(256)
rmi_bcepool_kernel(const float* __restrict__ logits,
                   const int*   __restrict__ labels,
                   float* __restrict__ ws) {
    __shared__ float sb[256];
    __shared__ float sv[256];
    int tid = threadIdx.x;
    int idx = blockIdx.x * 256 + tid;
    float bce = 0.0f, vcnt = 0.0f;
    if (idx < NCHAN * PH * PW) {
        int ow = idx % PW;
        int t  = idx / PW;
        int oh = t % PH;
        int nc = t / PH;
        int c  = nc % NC;
        int n  = nc / NC;
        float mla = -INFINITY, mpr = -INFINITY;
        for (int ky = 0; ky < 3; ky++) {
            int ih = 3*oh - 1 + ky;
            if (ih < 0 || ih >= H) continue;
            for (int kx = 0; kx < 3; kx++) {
                int iw = 3*ow - 1 + kx;
                if (iw < 0 || iw >= W) continue;
                int lab = labels[(n*H + ih)*W + iw];
                bool m  = lab < NC;
                float x = logits[((n*NC + c)*H + ih)*W + iw];
                float mf  = m ? 1.0f : 0.0f;
                float tgt = (m && lab == c) ? 1.0f : 0.0f;
                // stable BCE-with-logits, weighted by mask
                bce += (fmaxf(x, 0.0f) - x*tgt + log1pf(expf(-fabsf(x)))) * mf;
                if (c == 0) vcnt += mf;      // count each pixel exactly once
                float la = tgt;              // one-hot * mask
                float pr = (m ? 1.0f/(1.0f + expf(-x)) : 0.0f) + CLIP_MIN;
                mla = fmaxf(mla, la);
                mpr = fmaxf(mpr, pr);
            }
        }
        ws[WS_POOL_LA + idx] = mla;
        ws[WS_POOL_PR + idx] = mpr;
    }
    sb[tid] = bce; sv[tid] = vcnt;
    __syncthreads();
    for (int s = 128; s > 0; s >>= 1) {
        if (tid < s) { sb[tid] += sb[tid + s]; sv[tid] += sv[tid + s]; }
        __syncthreads();
    }
    if (tid == 0) { atomicAdd(&ws[0], sb[0]); atomicAdd(&ws[1], sv[0]); }
}

// -------------------------------- 9x9 raw Grams + row sums via V_WMMA_F32_16X16X4_F32
// One block per (n,c). Pooled planes are staged tile-by-tile into LDS with
// async global->LDS loads (coalesced, single read of each element), then the
// 8 waves run the WMMA rank-4 updates reading scattered data from LDS.
// A-fragment (16x4 f32): lane L<16 -> row L, K = {0,1}; lane L>=16 -> row L-16, K = {2,3}.
// For a Gram XY^T the B-fragment register contents equal the A-fragment of Y.
// Rows/cols 9..15 of the 16x16 D tile are never read, so lanes with row>=9
// may carry arbitrary values (no zero-guard needed; only the K tail is zeroed).
__global__ void __launch_bounds__(256) rmi_gram_kernel(float* __restrict__ ws) {
    __shared__ float lds_la[LDSN];
    __shared__ float lds_pr[LDSN];

    int nc = blockIdx.x;
    float* gla = ws + WS_POOL_LA + (size_t)nc * PH * PW;
    float* gpr = ws + WS_POOL_PR + (size_t)nc * PH * PW;
    float* G = ws + WS_G + nc * 3 * 81;
    float* S = ws + WS_S + nc * 2 * 9;

    int lane  = threadIdx.x & 31;
    int wave  = threadIdx.x >> 5;          // 0..7
    int row   = lane & 15;                 // Gram row/col index (valid if < 9)
    int kbase = (lane >> 4) << 1;          // 0 or 2
    bool rv   = row < 9;
    int dy = rv ? row / 3 : 0;             // crop shift (clamped for idle rows)
    int dx = rv ? row % 3 : 0;

    v8f acc_ll = {};
    v8f acc_pp = {};
    v8f acc_lp = {};
    float sla = 0.0f, spr = 0.0f;

    for (int y0 = 0; y0 < NH; y0 += TROWS) {
        int rows  = (NH - y0 < TROWS) ? (NH - y0) : TROWS;   // crop rows this tile
        int srows = rows + 2;                                 // pooled rows staged
        int total = srows * PW;
        float* sla_g = gla + y0 * PW;
        float* spr_g = gpr + y0 * PW;
#if USE_ASYNC_LDS
        for (int i = threadIdx.x; i < total; i += 256) {
            __builtin_amdgcn_global_load_async_to_lds_b32(
                (__attribute__((address_space(1))) int*)(sla_g + i),
                (__attribute__((address_space(3))) int*)(&lds_la[i]), 0, 0);
            __builtin_amdgcn_global_load_async_to_lds_b32(
                (__attribute__((address_space(1))) int*)(spr_g + i),
                (__attribute__((address_space(3))) int*)(&lds_pr[i]), 0, 0);
        }
#if __has_builtin(__builtin_amdgcn_s_wait_asynccnt)
        __builtin_amdgcn_s_wait_asynccnt(0);
#else
        asm volatile("s_wait_asynccnt 0x0" ::: "memory");
#endif
#else
        for (int i = threadIdx.x; i < total; i += 256) {
            lds_la[i] = sla_g[i];
            lds_pr[i] = spr_g[i];
        }
#endif
        __syncthreads();

        int mTile = rows * NW;             // contiguous m-range in this tile
        // lane's own K-slot position within the tile, tracked incrementally
        int ml = wave * 4 + kbase;         // < 32 < NW, so starts in row 0
        int ry = 0, mx = ml;
        for (int m0 = wave * 4; m0 < mTile; m0 += 32) {
            bool ok0 = (m0 + kbase)     < mTile;
            bool ok1 = (m0 + kbase + 1) < mTile;
            int mx1 = mx + 1, ry1 = ry;
            if (mx1 >= NW) { mx1 = 0; ry1++; }
            int off0 = (ry  + dy) * PW + (mx  + dx);
            int off1 = (ry1 + dy) * PW + (mx1 + dx);
            float l0 = lds_la[off0]; l0 = ok0 ? l0 : 0.0f;
            float l1 = lds_la[off1]; l1 = ok1 ? l1 : 0.0f;
            float p0 = lds_pr[off0]; p0 = ok0 ? p0 : 0.0f;
            float p1 = lds_pr[off1]; p1 = ok1 ? p1 : 0.0f;
            sla += l0 + l1;
            spr += p0 + p1;
            v2f fl = { l0, l1 };
            v2f fp = { p0, p1 };
            acc_ll = __builtin_amdgcn_wmma_f32_16x16x4_f32(false, fl, false, fl,
                                                           (short)0, acc_ll, false, false);
            acc_pp = __builtin_amdgcn_wmma_f32_16x16x4_f32(false, fp, false, fp,
                                                           (short)0, acc_pp, false, false);
            acc_lp = __builtin_amdgcn_wmma_f32_16x16x4_f32(false, fl, false, fp,
                                                           (short)0, acc_lp, false, false);
            mx += 32;
            if (mx >= NW) { mx -= NW; ry++; }
        }
        __syncthreads();   // protect LDS before next tile's staging
    }

    // Row sums: lane L and lane L+16 both hold partials of row L.
    if (rv) {
        atomicAdd(&S[row],     sla);
        atomicAdd(&S[9 + row], spr);
    }
    // D layout: lane<16 -> (M=v, N=lane); lane>=16 -> (M=v+8, N=lane-16).
    if (rv) {
        #pragma unroll
        for (int v = 0; v < 8; v++) {
            int Mrow = (lane < 16) ? v : v + 8;
            if (Mrow < 9) {
                atomicAdd(&G[      Mrow*9 + row], acc_ll[v]);
                atomicAdd(&G[ 81 + Mrow*9 + row], acc_pp[v]);
                atomicAdd(&G[162 + Mrow*9 + row], acc_lp[v]);
            }
        }
    }
}

// ---------------- per-(n,c): center, invert pr_cov, Schur complement, Cholesky logdet
__global__ void rmi_solve_kernel(float* __restrict__ ws) {
    int nc = blockIdx.x * blockDim.x + threadIdx.x;
    if (nc >= NCHAN) return;
    const float* G = ws + WS_G + nc * 3 * 81;
    const float* S = ws + WS_S + nc * 2 * 9;
    float cll[9][9], cpp[9][9], clp[9][9];
    const float invM = 1.0f / (float)MTOT;
    for (int i = 0; i < 9; i++)
        for (int j = 0; j < 9; j++) {
            cll[i][j] = G[      i*9 + j] - S[i]    * S[j]    * invM;
            cpp[i][j] = G[ 81 + i*9 + j] - S[9+i]  * S[9+j]  * invM;
            clp[i][j] = G[162 + i*9 + j] - S[i]    * S[9+j]  * invM;
        }
    for (int i = 0; i < 9; i++) cpp[i][i] += POS_ALPHA;

    // Gauss-Jordan inverse of cpp with partial pivoting
    float A[9][18];
    for (int i = 0; i < 9; i++)
        for (int j = 0; j < 9; j++) {
            A[i][j]     = cpp[i][j];
            A[i][9 + j] = (i == j) ? 1.0f : 0.0f;
        }
    for (int col = 0; col < 9; col++) {
        int p = col; float best = fabsf(A[col][col]);
        for (int r = col + 1; r < 9; r++) {
            float v = fabsf(A[r][col]);
            if (v > best) { best = v; p = r; }
        }
        if (p != col)
            for (int j = 0; j < 18; j++) { float t = A[col][j]; A[col][j] = A[p][j]; A[p][j] = t; }
        float d = 1.0f / A[col][col];
        for (int j = 0; j < 18; j++) A[col][j] *= d;
        for (int r = 0; r < 9; r++) {
            if (r == col) continue;
            float f = A[r][col];
            for (int j = 0; j < 18; j++) A[r][j] -= f * A[col][j];
        }
    }
    // T = clp @ inv(cpp)
    float T[9][9];
    for (int i = 0; i < 9; i++)
        for (int j = 0; j < 9; j++) {
            float s = 0.0f;
            for (int k = 0; k < 9; k++) s += clp[i][k] * A[k][9 + j];
            T[i][j] = s;
        }
    // appro_var = cll - T @ clp^T + alpha*I
    float ap[9][9];
    for (int i = 0; i < 9; i++)
        for (int j = 0; j < 9; j++) {
            float s = 0.0f;
            for (int k = 0; k < 9; k++) s += T[i][k] * clp[j][k];
            ap[i][j] = cll[i][j] - s;
        }
    for (int i = 0; i < 9; i++) ap[i][i] += POS_ALPHA;

    // Cholesky logdet
    float L[9][9];
    float logdet = 0.0f;
    for (int i = 0; i < 9; i++)
        for (int j = 0; j < 9; j++) L[i][j] = 0.0f;
    for (int i = 0; i < 9; i++) {
        for (int j = 0; j <= i; j++) {
            float s = ap[i][j];
            for (int k = 0; k < j; k++) s -= L[i][k] * L[j][k];
            if (j == i) {
                float dd = sqrtf(fmaxf(s, 1e-30f));
                L[i][i] = dd;
                logdet += logf(dd);
            } else {
                L[i][j] = s / L[j][j];
            }
        }
    }
    logdet *= 2.0f;
    float rmi_now = 0.5f * logdet;
    // rmi_loss contribution: mean over batch (4), /HALF_D (9)
    atomicAdd(&ws[2], rmi_now * (1.0f / 36.0f));
}

// ---------------------------------------------------------------- finalize
__global__ void rmi_final_kernel(const float* __restrict__ ws, float* __restrict__ out) {
    if (threadIdx.x == 0 && blockIdx.x == 0) {
        float bce = 0.5f * ws[0] / (ws[1] + 1.0f);  // lambda * bce_loss
        float rmi = 0.5f * ws[2];                   // (1-lambda) * rmi_loss
        out[0] = bce + rmi;
    }
}

extern "C" void kernel_launch(void* const* d_in, const int* in_sizes, int n_in,
                              void* d_out, int out_size, void* d_ws, size_t ws_size,
                              hipStream_t stream) {
    const float* logits = (const float*)d_in[0];
    const int*   labels = (const int*)d_in[1];
    float* ws  = (float*)d_ws;
    float* out = (float*)d_out;

    {
        int n = 16 + GSZ + SSZ;
        rmi_zero_kernel<<<(n + 255) / 256, 256, 0, stream>>>(ws);
    }
    {
        int n = NCHAN * PH * PW;
        rmi_bcepool_kernel<<<(n + 255) / 256, 256, 0, stream>>>(logits, labels, ws);
    }
    rmi_gram_kernel<<<NCHAN, 256, 0, stream>>>(ws);
    rmi_solve_kernel<<<1, 96, 0, stream>>>(ws);
    rmi_final_kernel<<<1, 32, 0, stream>>>(ws, out);
}